// TensplitGCNLARGE_74182675136540
// MI455X (gfx1250) — compile-verified
//
#include <hip/hip_runtime.h>

#define N_NODES 100000
#define IN_DIM  512
#define HIDDEN  128
#define OUT_DIM 64
#define N_EDGES 3200000

// K-pair-interleaved LDS layouts: element (k, n) lives at
//   (k>>1)*STRIDE + 2*n + (k&1)
// so a WMMA B fragment (values k0, k0+1 of column n) is one contiguous b64.
// STRIDE mod 64 == 32 puts the two half-waves (K offset 2 -> next pair-row)
// on disjoint bank halves.
#define W1P_STRIDE 288   // 64 pair-rows x 288 words = 72KB
#define W2P_STRIDE 160   // 64 pair-rows x 160 words = 40KB
#define H1_STRIDE  132   // h1 tile row stride: 132 mod 64 == 4 -> conflict-free b64

typedef float v2f __attribute__((ext_vector_type(2)));
typedef float v8f __attribute__((ext_vector_type(8)));

#if defined(__has_builtin)
#if __has_builtin(__builtin_amdgcn_global_load_async_to_lds_b32)
#define HAVE_ASYNC_LDS 1
#endif
#endif

__device__ __forceinline__ void g2l_async_b32(const float* g, float* l) {
#ifdef HAVE_ASYNC_LDS
    // Toolchain signature (from diagnostics): params are plain 'int*'.
    __builtin_amdgcn_global_load_async_to_lds_b32((int*)g, (int*)l, 0, 0);
#endif
}

__device__ __forceinline__ void wait_async_lds() {
#ifdef HAVE_ASYNC_LDS
    asm volatile("s_wait_asynccnt 0x0" ::: "memory");
#endif
}

// ---------------------------------------------------------------------------
// Fused  h2 = relu(X @ W1) @ W2   (all fp32, WMMA f32 16x16x4)
// Block = 256 threads = 8 waves; each wave computes a 16-row output tile.
// ---------------------------------------------------------------------------
__global__ __launch_bounds__(256) void gcn_fused_mlp(
    const float* __restrict__ X,
    const float* __restrict__ W1,
    const float* __restrict__ W2,
    float* __restrict__ H2)
{
    __shared__ __align__(16) float lds_w1[64 * W1P_STRIDE];  // 72KB (reused for h1)
    __shared__ __align__(16) float lds_w2[64 * W2P_STRIDE];  // 40KB

    const int tid  = threadIdx.x;
    const int wave = tid >> 5;
    const int lane = tid & 31;
    const int lo   = lane & 15;   // M (A) / N (B,C) index within half-wave
    const int hi   = lane >> 4;   // half-wave: K offset 2 (A/B), M offset 8 (C/D)
    const int row_base = (blockIdx.x * 8 + wave) * 16;

    // ---- stage W2 (128x64) into interleaved LDS layout ----
#ifdef HAVE_ASYNC_LDS
    for (int e = tid; e < HIDDEN * OUT_DIM; e += 256) {
        const int k = e >> 6, n = e & 63;
        g2l_async_b32(W2 + e, lds_w2 + (k >> 1) * W2P_STRIDE + 2 * n + (k & 1));
    }
#else
    for (int i = tid; i < HIDDEN * OUT_DIM / 4; i += 256) {
        const int q = i >> 5, p = i & 31;                 // pair-row, col-pair
        v2f ev = *(const v2f*)(W2 + (2 * q)     * OUT_DIM + 2 * p);
        v2f od = *(const v2f*)(W2 + (2 * q + 1) * OUT_DIM + 2 * p);
        float4 pk = make_float4(ev.x, od.x, ev.y, od.y);
        *(float4*)(lds_w2 + q * W2P_STRIDE + 4 * p) = pk;
    }
#endif

    v8f c1[8] = {};  // 16x128 fp32 accumulator (8 N-tiles of 16x16)

    const int   arow = (row_base + lo < N_NODES) ? (row_base + lo) : (N_NODES - 1);
    const float* aptr = X + (size_t)arow * IN_DIM + 2 * hi;     // A: M=lo, K pair @2*hi
    const int   bbase  = hi * W1P_STRIDE + 2 * lo;              // B frag base (W1)

    for (int chunk = 0; chunk < IN_DIM / 128; ++chunk) {
        __syncthreads();  // previous chunk fully consumed
        // ---- stage 128xHIDDEN chunk of W1 into interleaved layout ----
        const float* src = W1 + (size_t)chunk * 128 * HIDDEN;
#ifdef HAVE_ASYNC_LDS
        for (int e = tid; e < 128 * HIDDEN; e += 256) {
            const int k = e >> 7, n = e & 127;
            g2l_async_b32(src + e, lds_w1 + (k >> 1) * W1P_STRIDE + 2 * n + (k & 1));
        }
        wait_async_lds();
#else
        for (int i = tid; i < 128 * HIDDEN / 4; i += 256) {
            const int q = i >> 6, p = i & 63;             // pair-row, col-pair
            v2f ev = *(const v2f*)(src + (2 * q)     * HIDDEN + 2 * p);
            v2f od = *(const v2f*)(src + (2 * q + 1) * HIDDEN + 2 * p);
            float4 pk = make_float4(ev.x, od.x, ev.y, od.y);
            *(float4*)(lds_w1 + q * W1P_STRIDE + 4 * p) = pk;
        }
#endif
        __syncthreads();

        for (int ks = 0; ks < 32; ++ks) {                 // K step of 4
            v2f a = *(const v2f*)(aptr + chunk * 128 + ks * 4);
            const float* brow = lds_w1 + bbase + ks * (2 * W1P_STRIDE);
            #pragma unroll
            for (int t = 0; t < 8; ++t) {                 // 8 N-tiles over HIDDEN
                v2f b = *(const v2f*)(brow + t * 32);     // contiguous b64 fragment
                c1[t] = __builtin_amdgcn_wmma_f32_16x16x4_f32(
                    false, a, false, b, (short)0, c1[t], false, false);
            }
        }
    }
    __syncthreads();  // all waves done reading W1 chunks -> reuse lds_w1 for h1

    // ---- relu + park 16x128 h1 tile in this wave's LDS region ----
    float* hw = lds_w1 + wave * (16 * H1_STRIDE);
    #pragma unroll
    for (int t = 0; t < 8; ++t) {
        #pragma unroll
        for (int v = 0; v < 8; ++v) {
            // C/D layout: VGPR v -> row v + 8*hi, col = t*16 + lo
            hw[(v + 8 * hi) * H1_STRIDE + t * 16 + lo] = fmaxf(c1[t][v], 0.f);
        }
    }
    // same-wave LDS ops are in-order: safe to read back without a barrier

    // ---- GEMM2: 16x128 (LDS) @ 128x64 (LDS) -> 16x64 ----
    v8f c2[4] = {};
    const float* a2p    = hw + lo * H1_STRIDE + 2 * hi;
    const int    b2base = hi * W2P_STRIDE + 2 * lo;
    for (int ks = 0; ks < HIDDEN / 4; ++ks) {
        v2f a = *(const v2f*)(a2p + ks * 4);
        const float* brow = lds_w2 + b2base + ks * (2 * W2P_STRIDE);
        #pragma unroll
        for (int t = 0; t < 4; ++t) {
            v2f b = *(const v2f*)(brow + t * 32);
            c2[t] = __builtin_amdgcn_wmma_f32_16x16x4_f32(
                false, a, false, b, (short)0, c2[t], false, false);
        }
    }

    // ---- store 16x64 tile ----
    #pragma unroll
    for (int t = 0; t < 4; ++t) {
        #pragma unroll
        for (int v = 0; v < 8; ++v) {
            const int r = row_base + v + 8 * hi;
            if (r < N_NODES)
                H2[(size_t)r * OUT_DIM + t * 16 + lo] = c2[t][v];
        }
    }
}

// ---------------------------------------------------------------------------
// One COO propagation: out[dst] += w * h[src]  (64 cols; one wave per edge).
// h (25.6MB) and out (25.6MB) both fit in the 192MB L2: gathers hit L2 and
// the f32 adds resolve as L2 atomics (global_atomic_add_f32).
// ---------------------------------------------------------------------------
__global__ __launch_bounds__(256) void gcn_propagate(
    const float* __restrict__ h,
    const float* __restrict__ ew,
    const int*   __restrict__ esrc,
    const int*   __restrict__ edst,
    float*       __restrict__ out)
{
    const int e    = (int)((blockIdx.x * 256u + threadIdx.x) >> 5);
    const int lane = threadIdx.x & 31;
    if (e >= N_EDGES) return;

    const int   s = esrc[e];
    const int   d = edst[e];
    const float w = ew[e];

    const float* hp = h   + (size_t)s * OUT_DIM;
    float*       op = out + (size_t)d * OUT_DIM;

    const float m0 = w * hp[lane];
    const float m1 = w * hp[lane + 32];
    unsafeAtomicAdd(op + lane,      m0);   // -> global_atomic_add_f32
    unsafeAtomicAdd(op + lane + 32, m1);
}

// ---------------------------------------------------------------------------
extern "C" void kernel_launch(void* const* d_in, const int* in_sizes, int n_in,
                              void* d_out, int out_size, void* d_ws, size_t ws_size,
                              hipStream_t stream) {
    const float* features = (const float*)d_in[0];   // [N, 512]
    const float* W1       = (const float*)d_in[1];   // [512, 128]
    const float* W2       = (const float*)d_in[2];   // [128, 64]
    const float* ew       = (const float*)d_in[3];   // [E]
    const int*   esrc     = (const int*)d_in[4];     // [E]
    const int*   edst     = (const int*)d_in[5];     // [E]
    float*       out      = (float*)d_out;           // [N, 64]

    const size_t hElems = (size_t)N_NODES * OUT_DIM;
    float* h2 = (float*)d_ws;          // relu(X@W1)@W2
    float* z1 = h2 + hElems;           // first propagation result

    // zero the accumulation buffers every launch (graph-capturable memsets)
    (void)hipMemsetAsync(z1,  0, hElems * sizeof(float), stream);
    (void)hipMemsetAsync(out, 0, hElems * sizeof(float), stream);

    const int mlpBlocks  = (N_NODES + 127) / 128;    // 8 waves x 16 rows per block
    gcn_fused_mlp<<<mlpBlocks, 256, 0, stream>>>(features, W1, W2, h2);

    const int propBlocks = N_EDGES / 8;              // 8 edges (waves) per block
    gcn_propagate<<<propBlocks, 256, 0, stream>>>(h2, ew, esrc, edst, z1);
    gcn_propagate<<<propBlocks, 256, 0, stream>>>(z1, ew, esrc, edst, out);
}